// CodeAwareGAT_4690104287576
// MI455X (gfx1250) — compile-verified
//
#include <hip/hip_runtime.h>
#include <hip/hip_bf16.h>

typedef __attribute__((ext_vector_type(2))) float v2f;
typedef __attribute__((ext_vector_type(8))) float v8f;
typedef unsigned int u32x4 __attribute__((ext_vector_type(4)));
typedef unsigned int u32x8 __attribute__((ext_vector_type(8)));

#define NEG_SLOPE 0.2f
#define LN_EPS 1e-6f

// ---------- helpers ----------

__device__ __forceinline__ float wave_sum(float v) {
#pragma unroll
  for (int off = 16; off > 0; off >>= 1) v += __shfl_xor(v, off, 32);
  return v;
}

// order-preserving float<->uint mapping for atomicMax on floats (any sign)
__device__ __forceinline__ unsigned f32_ord(float f) {
  unsigned u = __float_as_uint(f);
  return (u & 0x80000000u) ? ~u : (u | 0x80000000u);
}
__device__ __forceinline__ float ord_f32(unsigned u) {
  return (u & 0x80000000u) ? __uint_as_float(u & 0x7fffffffu)
                           : __uint_as_float(~u);
}

__device__ __forceinline__ void edge_sd(const int* ei, int E, int e, int& s, int& d) {
  if (e < E) { s = ei[e]; d = ei[E + e]; }      // edge_index is (2,E) row-major
  else       { s = e - E; d = e - E; }          // appended self loops
}

// ---------- Tensor Data Mover: 2D tile Global -> LDS ----------
// Builds the D# descriptor (groups 0/1) per CDNA5 ISA ch.8 and issues
// tensor_load_to_lds via inline asm (wave-uniform; tracked by TENSORcnt).
//   tile_w  : elements per row (contiguous, 4B each)
//   tile_h  : rows
//   stride  : row stride of the global tensor, in elements
//   pad     : group1 LDS-padding control bits (or 0)

#define TDM_PAD_64x2 ((1u << 20) | (5u << 22) | (1u << 25)) // +2 dwords every 64 dwords

__device__ __forceinline__ void tdm_load_2d(const void* gsrc, unsigned lds_off,
                                            unsigned tile_w, unsigned tile_h,
                                            unsigned long stride, unsigned pad) {
  const unsigned long ga = (unsigned long)gsrc;
  const unsigned TD = 1u << 30;   // tensor dims: huge (tiles are always in-bounds)
  u32x4 g0;
  g0[0] = 1u;                                               // count=1, user D#
  g0[1] = lds_off;                                          // lds_addr (bytes)
  g0[2] = (unsigned)ga;                                     // global_addr[31:0]
  g0[3] = ((unsigned)(ga >> 32) & 0x01FFFFFFu) | (2u << 30);// ga[56:32] | type=2
  u32x8 g1;
  g1[0] = (2u << 16) | pad;                                 // data_size=4B, pad ctl
  g1[1] = (TD & 0xFFFFu) << 16;                             // tensor_dim0[15:0]
  g1[2] = ((TD >> 16) & 0xFFFFu) | ((TD & 0xFFFFu) << 16);  // d0 hi | d1 lo
  g1[3] = ((TD >> 16) & 0xFFFFu) | ((tile_w & 0xFFFFu) << 16); // d1 hi | tile_dim0
  g1[4] = tile_h & 0xFFFFu;                                 // tile_dim1 (dim2=0)
  g1[5] = (unsigned)stride;                                 // dim0_stride[31:0]
  g1[6] = (unsigned)(stride >> 32) & 0xFFFFu;               // stride hi | d1_stride=0
  g1[7] = 0u;
  asm volatile("tensor_load_to_lds %0, %1" :: "s"(g0), "s"(g1) : "memory");
}

// ---------- fill ----------

__global__ void fill_f32_k(float* p, float v, long n) {
  long i = (long)blockIdx.x * blockDim.x + threadIdx.x;
  if (i < n) p[i] = v;
}
__global__ void fill_u32_k(unsigned* p, unsigned v, long n) {
  long i = (long)blockIdx.x * blockDim.x + threadIdx.x;
  if (i < n) p[i] = v;
}

// ---------- WMMA f32 GEMM with TDM double-buffered LDS staging ----------
// Block = 4 waves = 16 rows x 64 cols of C. K is chunked by KT=64.
// Wave 0 TDM-loads the shared 16xKT A chunk (with LDS padding: row stride 66
// dwords -> conflict-free banks); wave 1 TDM-loads the KTx64 B chunk shared by
// all four waves. Double buffered; one workgroup barrier per chunk.
// V_WMMA_F32_16X16X4_F32 gives exact f32 GEMM math on the matrix pipe.

#define KT   64
#define AROW 66   // 64 data dwords + 2 TDM pad dwords per row

__global__ __launch_bounds__(128)
void gemm_wmma_f32_tdm(const float* __restrict__ A, long lda,
                       const float* __restrict__ B, long ldb,
                       float* __restrict__ C, long ldc, int K) {
  __shared__ float lA[2][16 * AROW];
  __shared__ float lB[2][KT * 64];

  const int wave = threadIdx.x >> 5;
  const int lane = threadIdx.x & 31;
  const int half = lane >> 4;       // 0: K pair {0,1}, 1: K pair {2,3}
  const int l    = lane & 15;
  const long row0    = (long)blockIdx.x * 16;
  const long col0blk = (long)blockIdx.y * 64;
  const int nc = K / KT;

  if (wave == 0)
    tdm_load_2d(A + row0 * lda, (unsigned)(uintptr_t)&lA[0][0],
                KT, 16, (unsigned long)lda, TDM_PAD_64x2);
  if (wave == 1)
    tdm_load_2d(B + col0blk, (unsigned)(uintptr_t)&lB[0][0],
                64, KT, (unsigned long)ldb, 0u);

  v8f acc = {};
  for (int c = 0; c < nc; ++c) {
    const int p = c & 1;
    __builtin_amdgcn_s_wait_tensorcnt(0);   // issuing waves: chunk c landed
    __syncthreads();                        // publish LDS to all waves

    if (c + 1 < nc) {                       // prefetch next chunk into other buf
      const long k0n = (long)(c + 1) * KT;
      if (wave == 0)
        tdm_load_2d(A + row0 * lda + k0n, (unsigned)(uintptr_t)&lA[1 - p][0],
                    KT, 16, (unsigned long)lda, TDM_PAD_64x2);
      if (wave == 1)
        tdm_load_2d(B + k0n * ldb + col0blk, (unsigned)(uintptr_t)&lB[1 - p][0],
                    64, KT, (unsigned long)ldb, 0u);
    }

    const float* abase = &lA[p][l * AROW + 2 * half];
    const float* bbase = &lB[p][half * 2 * 64 + wave * 16 + l];
#pragma unroll
    for (int kk = 0; kk < KT; kk += 4) {
      v2f a; a.x = abase[kk];       a.y = abase[kk + 1];
      v2f b; b.x = bbase[kk * 64];  b.y = bbase[kk * 64 + 64];
      acc = __builtin_amdgcn_wmma_f32_16x16x4_f32(
          /*neg_a=*/false, a, /*neg_b=*/false, b,
          /*c_mod=*/(short)0, acc, /*reuse_a=*/false, /*reuse_b=*/false);
    }
  }

  // C layout: VGPR i -> rows (i, i+8), lanes 16-31 rows +8
  const long col0 = col0blk + (long)wave * 16;
  float* Crow = C + (row0 + half * 8) * ldc + col0 + l;
#pragma unroll
  for (int i = 0; i < 8; ++i) Crow[(long)i * ldc] = acc[i];
}

// ---------- attention dot products: a_src[n,h] = <feat[n,h,:], att_src[h,:]> ----------

__global__ void att_dots_k(const float* __restrict__ feat,
                           const float* __restrict__ att_src,
                           const float* __restrict__ att_dst,
                           float* __restrict__ a_src, float* __restrict__ a_dst,
                           int heads, int C) {
  const int n = blockIdx.x;
  const int hd = threadIdx.x >> 5;
  const int lane = threadIdx.x & 31;
  if (hd >= heads) return;
  const float* fp = feat + ((long)n * heads + hd) * C;
  const float* as = att_src + (long)hd * C;
  const float* ad = att_dst + (long)hd * C;
  float ss = 0.f, ds = 0.f;
  for (int c = lane; c < C; c += 32) {
    float v = fp[c];
    ss += v * as[c];
    ds += v * ad[c];
  }
  ss = wave_sum(ss);
  ds = wave_sum(ds);
  if (lane == 0) {
    a_src[(long)n * heads + hd] = ss;
    a_dst[(long)n * heads + hd] = ds;
  }
}

// ---------- edge softmax: logits + segment max ----------

__global__ void edge_logits_k(const int* __restrict__ ei, int E, int Etot, int heads,
                              const float* __restrict__ a_src,
                              const float* __restrict__ a_dst,
                              float* __restrict__ ebuf, unsigned* __restrict__ mord) {
  long idx = (long)blockIdx.x * blockDim.x + threadIdx.x;
  if (idx >= (long)Etot * heads) return;
  int e = (int)(idx / heads), h = (int)(idx % heads);
  int s, d; edge_sd(ei, E, e, s, d);
  float v = a_src[(long)s * heads + h] + a_dst[(long)d * heads + h];
  v = v > 0.f ? v : NEG_SLOPE * v;                 // leaky_relu
  ebuf[idx] = v;
  atomicMax(&mord[(long)d * heads + h], f32_ord(v));
}

// ---------- edge softmax: exp + segment sum ----------

__global__ void edge_exp_k(const int* __restrict__ ei, int E, int Etot, int heads,
                           const unsigned* __restrict__ mord,
                           float* __restrict__ ebuf, float* __restrict__ ssum) {
  long idx = (long)blockIdx.x * blockDim.x + threadIdx.x;
  if (idx >= (long)Etot * heads) return;
  int e = (int)(idx / heads), h = (int)(idx % heads);
  int s, d; edge_sd(ei, E, e, s, d);
  float m = ord_f32(mord[(long)d * heads + h]);
  float ex = __expf(ebuf[idx] - m);
  ebuf[idx] = ex;
  atomicAdd(&ssum[(long)d * heads + h], ex);
}

// ---------- message aggregation: out[dst,h,:] += alpha * feat[src,h,:] ----------

__global__ __launch_bounds__(256)
void aggregate_k(const int* __restrict__ ei, int E, int heads, int C,
                 const float* __restrict__ ebuf, const float* __restrict__ ssum,
                 const float* __restrict__ feat, float* __restrict__ out,
                 float* __restrict__ alpha_out) {
  const int e = blockIdx.x;
  int s, d; edge_sd(ei, E, e, s, d);
  // warm the gathered source row (lowers to global_prefetch_b8)
  __builtin_prefetch(feat + (long)s * heads * C + threadIdx.x, 0, 3);
  __shared__ float sal[8];
  if ((int)threadIdx.x < heads) {
    float al = ebuf[(long)e * heads + threadIdx.x] /
               ssum[(long)d * heads + threadIdx.x];
    sal[threadIdx.x] = al;
    if (alpha_out) alpha_out[(long)e * heads + threadIdx.x] = al;
  }
  __syncthreads();
  for (int h = 0; h < heads; ++h) {
    const float al = sal[h];
    const float* fp = feat + ((long)s * heads + h) * C;
    float* op = out + ((long)d * heads + h) * C;
    for (int c = threadIdx.x; c < C; c += blockDim.x)
      atomicAdd(&op[c], al * fp[c]);
  }
}

// ---------- bias (+optional relu) ----------

__global__ void bias_act_k(float* __restrict__ g, const float* __restrict__ bias,
                           long n, int width, int do_relu) {
  long i = (long)blockIdx.x * blockDim.x + threadIdx.x;
  if (i >= n) return;
  float v = g[i] + bias[i % width];
  g[i] = do_relu ? fmaxf(v, 0.f) : v;
}

// ---------- finalize: y = x + g2 ; out = ln_w * y * rsqrt(mean(y*y)+eps) ----------

__global__ __launch_bounds__(256)
void finalize_k(const float* __restrict__ x, const float* __restrict__ g2,
                const float* __restrict__ ln_w, float* __restrict__ out,
                int S, int H) {
  const long row = blockIdx.x;          // n*S + s
  const long n = row / S;
  const float* xr = x + row * H;
  const float* gr = g2 + n * H;
  const int c = threadIdx.x;            // H == blockDim.x == 256
  float y = xr[c] + gr[c];

  __shared__ float red[8];
  float p = wave_sum(y * y);
  if ((threadIdx.x & 31) == 0) red[threadIdx.x >> 5] = p;
  __syncthreads();
  float tot = 0.f;
#pragma unroll
  for (int i = 0; i < 8; ++i) tot += red[i];
  float var = tot / (float)H;
  out[row * H + c] = ln_w[c] * y * rsqrtf(var + LN_EPS);
}

// ---------- driver ----------

static inline int cdiv(long a, int b) { return (int)((a + b - 1) / b); }

extern "C" void kernel_launch(void* const* d_in, const int* in_sizes, int n_in,
                              void* d_out, int out_size, void* d_ws, size_t ws_size,
                              hipStream_t stream) {
  const float* x        = (const float*)d_in[0];
  const int*   ei       = (const int*)d_in[1];
  const float* W1       = (const float*)d_in[2];
  const float* att_src1 = (const float*)d_in[3];
  const float* att_dst1 = (const float*)d_in[4];
  const float* bias1    = (const float*)d_in[5];
  const float* W2       = (const float*)d_in[6];
  const float* att_src2 = (const float*)d_in[7];
  const float* att_dst2 = (const float*)d_in[8];
  const float* bias2    = (const float*)d_in[9];
  const float* ln_w     = (const float*)d_in[10];

  const int H      = in_sizes[10];          // 256
  const int headsH = in_sizes[5];           // 1024
  const int heads  = headsH / H;            // 4
  const int E      = in_sizes[1] / 2;       // 320000
  const int S      = 32;
  const int N      = (int)((long)in_sizes[0] / ((long)S * H));  // 20000
  const int Etot   = E + N;                 // self loops appended

  // workspace layout (floats)
  float* h1   = (float*)d_ws;                         // N*headsH
  float* g1   = h1 + (long)N * headsH;                // N*headsH
  float* ebuf = g1 + (long)N * headsH;                // Etot*heads
  float* a_src = ebuf + (long)Etot * heads;           // N*heads
  float* a_dst = a_src + (long)N * heads;             // N*heads
  float* ssum  = a_dst + (long)N * heads;             // N*heads
  unsigned* mord = (unsigned*)(ssum + (long)N * heads); // N*heads
  // layer-2 buffers alias h1's region (h1 dead once g1 is built)
  float* h2 = h1;                                     // N*H
  float* g2 = h1 + (long)N * H;                       // N*H

  float* out_y     = (float*)d_out;                   // N*S*H
  float* alpha_out = out_y + (long)N * S * H;         // Etot (heads2 == 1)

  // ===== layer 1 =====
  fill_f32_k<<<cdiv((long)N * headsH, 256), 256, 0, stream>>>(g1, 0.f, (long)N * headsH);
  fill_u32_k<<<cdiv((long)N * heads, 256), 256, 0, stream>>>(mord, 0u, (long)N * heads);
  fill_f32_k<<<cdiv((long)N * heads, 256), 256, 0, stream>>>(ssum, 0.f, (long)N * heads);

  // h1 = node_feat @ W1 ; node_feat rows are x[:,0,:] -> lda = S*H
  gemm_wmma_f32_tdm<<<dim3(N / 16, headsH / 64), 128, 0, stream>>>(
      x, (long)S * H, W1, headsH, h1, headsH, H);

  att_dots_k<<<N, heads * 32, 0, stream>>>(h1, att_src1, att_dst1, a_src, a_dst, heads, H);

  edge_logits_k<<<cdiv((long)Etot * heads, 256), 256, 0, stream>>>(
      ei, E, Etot, heads, a_src, a_dst, ebuf, mord);
  edge_exp_k<<<cdiv((long)Etot * heads, 256), 256, 0, stream>>>(
      ei, E, Etot, heads, mord, ebuf, ssum);
  aggregate_k<<<Etot, 256, 0, stream>>>(ei, E, heads, H, ebuf, ssum, h1, g1, nullptr);

  bias_act_k<<<cdiv((long)N * headsH, 256), 256, 0, stream>>>(
      g1, bias1, (long)N * headsH, headsH, /*relu=*/1);

  // ===== layer 2 (heads = 1) =====
  fill_f32_k<<<cdiv((long)N * H, 256), 256, 0, stream>>>(g2, 0.f, (long)N * H);
  fill_u32_k<<<cdiv((long)N, 256), 256, 0, stream>>>(mord, 0u, (long)N);
  fill_f32_k<<<cdiv((long)N, 256), 256, 0, stream>>>(ssum, 0.f, (long)N);

  // h2 = g1 @ W2
  gemm_wmma_f32_tdm<<<dim3(N / 16, H / 64), 128, 0, stream>>>(
      g1, headsH, W2, H, h2, H, headsH);

  att_dots_k<<<N, 32, 0, stream>>>(h2, att_src2, att_dst2, a_src, a_dst, 1, H);

  edge_logits_k<<<cdiv(Etot, 256), 256, 0, stream>>>(
      ei, E, Etot, 1, a_src, a_dst, ebuf, mord);
  edge_exp_k<<<cdiv(Etot, 256), 256, 0, stream>>>(
      ei, E, Etot, 1, mord, ebuf, ssum);
  aggregate_k<<<Etot, 256, 0, stream>>>(ei, E, 1, H, ebuf, ssum, h2, g2, alpha_out);

  bias_act_k<<<cdiv((long)N * H, 256), 256, 0, stream>>>(
      g2, bias2, (long)N * H, H, /*relu=*/0);

  // ===== finalize =====
  finalize_k<<<(long)N * S, H, 0, stream>>>(x, g2, ln_w, out_y, S, H);
}